// BernsteinConv_42502996361881
// MI455X (gfx1250) — compile-verified
//
#include <hip/hip_runtime.h>
#include <hip/hip_bf16.h>
#include <cstdint>

// ---------------------------------------------------------------------------
// BernsteinConv (K=1, N_ORD=2, COMB=2):
//   deg[i]   = #edges with dst==i
//   dinv[i]  = clip(deg,1)^-0.5
//   sfeat    = feat * dinv                       (per node row)
//   agg[i]   = sum_{e: dst==i} sfeat[src_e]      (sparse scatter-add)
//   y        = feat - agg * dinv
//   out      = y * (feat - 0.5*y)                (== COMB * (y/2) * (feat - y/2))
//
// Working set (~52 MB) fits the 192 MB L2 -> edge pass is L2 atomic/latency
// bound. Strategy: wave32 lane == feature (D_FEAT == 32), 12500 independent
// waves, per-wave index slice staged via async global->LDS b128 copies.
// ---------------------------------------------------------------------------

#define D_FEAT 32

__device__ __forceinline__ void atomAddF32(float* p, float v) {
  // Lowers to global_atomic_add_f32 (no CAS loop) on gfx1250.
  __hip_atomic_fetch_add(p, v, __ATOMIC_RELAXED, __HIP_MEMORY_SCOPE_AGENT);
}

__device__ __forceinline__ unsigned lds_off(const void* p) {
  // Generic LDS address: low 32 bits are the LDS byte offset within the
  // workgroup allocation (aperture tag lives in the high bits).
  return (unsigned)(uintptr_t)p;
}

// --------------------------- kernels ---------------------------------------

__global__ void k_zero_f32(float* __restrict__ p, int n) {
  int i = blockIdx.x * blockDim.x + threadIdx.x;
  if (i < n) p[i] = 0.0f;
}

__global__ void k_degree(const int* __restrict__ edst, unsigned* __restrict__ deg,
                         int n_edges) {
  int i = (blockIdx.x * blockDim.x + threadIdx.x) * 4;
  if (i + 3 < n_edges) {
    int4 v = *(const int4*)(edst + i);
    __hip_atomic_fetch_add(&deg[v.x], 1u, __ATOMIC_RELAXED, __HIP_MEMORY_SCOPE_AGENT);
    __hip_atomic_fetch_add(&deg[v.y], 1u, __ATOMIC_RELAXED, __HIP_MEMORY_SCOPE_AGENT);
    __hip_atomic_fetch_add(&deg[v.z], 1u, __ATOMIC_RELAXED, __HIP_MEMORY_SCOPE_AGENT);
    __hip_atomic_fetch_add(&deg[v.w], 1u, __ATOMIC_RELAXED, __HIP_MEMORY_SCOPE_AGENT);
  } else {
    for (; i < n_edges; ++i)
      __hip_atomic_fetch_add(&deg[edst[i]], 1u, __ATOMIC_RELAXED,
                             __HIP_MEMORY_SCOPE_AGENT);
  }
}

__global__ void k_scale(const float* __restrict__ feat, const unsigned* __restrict__ deg,
                        float* __restrict__ sfeat, float* __restrict__ dinv, int n_feat) {
  int i = blockIdx.x * blockDim.x + threadIdx.x;
  if (i >= n_feat) return;
  int node = i >> 5;
  float dg = (float)deg[node];
  if (dg < 1.0f) dg = 1.0f;
  float di = 1.0f / sqrtf(dg);
  sfeat[i] = feat[i] * di;
  if ((i & 31) == 0) dinv[node] = di;
}

// Edge scatter: one wave per 128-edge slice; lane = feature index.
// Each wave stages its 128 src + 128 dst indices with TWO async b128
// global->LDS copies (32 lanes x 16B each), waits on ASYNCcnt, then streams
// gather (sfeat[src]) -> fp32 atomic scatter (agg[dst]), prefetching the
// next gather row.
#define WPB  8     // waves per block
#define EPW  128   // edges per wave

__global__ __launch_bounds__(WPB * 32)
void k_edge_scatter(const int* __restrict__ esrc, const int* __restrict__ edst,
                    const float* __restrict__ sfeat, float* __restrict__ agg,
                    int n_edges) {
  __shared__ int sidx[WPB][2][EPW];   // [wave][src/dst][edge]

  const int lane = threadIdx.x & 31;
  const int w    = threadIdx.x >> 5;
  const long long base = ((long long)blockIdx.x * WPB + w) * (long long)EPW;
  if (base >= n_edges) return;        // wave-uniform exit

  // Stage indices: lane L copies edges [base+4L, base+4L+3] of each array.
  long long idx  = base + (long long)lane * 4;
  long long maxi = (long long)n_edges - 4;           // clamp keeps b128 in bounds
  int off = (int)((idx > maxi ? maxi : idx) * 4);
  unsigned la_s = lds_off(&sidx[w][0][lane * 4]);
  unsigned la_d = lds_off(&sidx[w][1][lane * 4]);
  asm volatile("global_load_async_to_lds_b128 %0, %1, %2"
               :: "v"(la_s), "v"(off), "s"(esrc) : "memory");
  asm volatile("global_load_async_to_lds_b128 %0, %1, %2"
               :: "v"(la_d), "v"(off), "s"(edst) : "memory");
  asm volatile("s_wait_asynccnt 0x0" ::: "memory");

  long long rem = (long long)n_edges - base;
  const int count = (rem < EPW) ? (int)rem : EPW;

  #pragma unroll 4
  for (int e = 0; e < count; ++e) {
    int s = sidx[w][0][e];
    int d = sidx[w][1][e];
    if (e + 1 < count) {
      int sn = sidx[w][0][e + 1];
      __builtin_prefetch(&sfeat[(size_t)sn * D_FEAT + lane], 0, 0);
    }
    float v = sfeat[(size_t)s * D_FEAT + lane];
    atomAddF32(&agg[(size_t)d * D_FEAT + lane], v);
  }
}

__global__ void k_finish(const float* __restrict__ feat, const float* __restrict__ agg,
                         const float* __restrict__ dinv, float* __restrict__ out,
                         int n_feat) {
  int i = blockIdx.x * blockDim.x + threadIdx.x;
  if (i >= n_feat) return;
  int node = i >> 5;
  float f = feat[i];
  float y = f - agg[i] * dinv[node];
  out[i] = y * (f - 0.5f * y);
}

// --------------------------- launch ----------------------------------------

extern "C" void kernel_launch(void* const* d_in, const int* in_sizes, int n_in,
                              void* d_out, int out_size, void* d_ws, size_t ws_size,
                              hipStream_t stream) {
  const float* feat = (const float*)d_in[0];
  const int*   esrc = (const int*)d_in[1];
  const int*   edst = (const int*)d_in[2];
  float*       out  = (float*)d_out;

  const int n_feat  = in_sizes[0];            // N_NODES * 32
  const int n_nodes = n_feat / D_FEAT;
  const int n_edges = in_sizes[1];

  // workspace layout
  char* ws = (char*)d_ws;
  float*    agg   = (float*)ws;                               // n_feat floats
  float*    sfeat = agg + (size_t)n_feat;                     // n_feat floats
  unsigned* deg   = (unsigned*)(sfeat + (size_t)n_feat);      // n_nodes u32
  float*    dinv  = (float*)(deg + (size_t)n_nodes);          // n_nodes floats

  const int B = 256;

  // zero agg + deg
  k_zero_f32<<<(n_feat + B - 1) / B, B, 0, stream>>>(agg, n_feat);
  k_zero_f32<<<(n_nodes + B - 1) / B, B, 0, stream>>>((float*)deg, n_nodes);

  // degree histogram (int4-vectorized)
  int deg_threads = (n_edges + 3) / 4;
  k_degree<<<(deg_threads + B - 1) / B, B, 0, stream>>>(edst, deg, n_edges);

  // per-node scale
  k_scale<<<(n_feat + B - 1) / B, B, 0, stream>>>(feat, deg, sfeat, dinv, n_feat);

  // sparse scatter-add (async b128-staged indices, fp32 atomics)
  long long waves  = ((long long)n_edges + EPW - 1) / EPW;
  int       blocks = (int)((waves + WPB - 1) / WPB);
  k_edge_scatter<<<blocks, WPB * 32, 0, stream>>>(esrc, edst, sfeat, agg, n_edges);

  // elementwise finish
  k_finish<<<(n_feat + B - 1) / B, B, 0, stream>>>(feat, agg, dinv, out, n_feat);
}